// GMConv_73229192397135
// MI455X (gfx1250) — compile-verified
//
#include <hip/hip_runtime.h>
#include <hip/hip_bf16.h>

// ---------------------------------------------------------------------------
// Problem constants (from reference)
// ---------------------------------------------------------------------------
constexpr int NI = 8192;   // inputs
constexpr int NK = 64;     // kernels
constexpr int NC = 16;     // components per kernel
constexpr int NF = 32;     // feature dim
constexpr int HD = 32;     // MLP hidden dim

// output layout (concatenated flat, reference return order)
constexpr int OFF_W = 0;           // w_out:   NI
constexpr int OFF_M = NI;          // m_out:   NI*3
constexpr int OFF_C = NI * 4;      // c_out:   NI*9
constexpr int OFF_F = NI * 13;     // features:NI*32

// workspace layout (floats)
//  ws[0..31]  : v0w  = S @ W0_w            (w-MLP folded first layer)
//  ws[32]     : alpha_m
//  ws[33..35] : C_m[3]
//  ws[36]     : alpha_c
//  ws[37..45] : C_c[9]

typedef __attribute__((ext_vector_type(2))) float v2f;
typedef __attribute__((ext_vector_type(8))) float v8f;

// ---------------------------------------------------------------------------
// Precompute: per-kernel merged stats + folded affine MLPs (1 block, 64 thr)
// ---------------------------------------------------------------------------
__global__ __launch_bounds__(64) void gm_precompute(
    const float* __restrict__ kw,   // (NK,NC)
    const float* __restrict__ km,   // (NK,NC,3)
    const float* __restrict__ ksc,  // (NK,NC,3)
    const float* __restrict__ kq,   // (NK,NC,4)
    const float* __restrict__ W0w,  // (64,32)  w-MLP layer 0
    const float* __restrict__ W0m, const float* __restrict__ b0m,
    const float* __restrict__ W1m, const float* __restrict__ b1m,
    const float* __restrict__ W2m, const float* __restrict__ b2m,
    const float* __restrict__ W3m, const float* __restrict__ b3m,
    const float* __restrict__ W0c, const float* __restrict__ b0c,
    const float* __restrict__ W1c, const float* __restrict__ b1c,
    const float* __restrict__ W2c, const float* __restrict__ b2c,
    const float* __restrict__ W3c, const float* __restrict__ b3c,
    float* __restrict__ ws)
{
    __shared__ float S[NK];
    __shared__ float KM[NK][3];
    __shared__ float KC[NK][9];
    __shared__ float u2[HD], u1[HD], veff[NK];

    const int tid = threadIdx.x;   // 0..63, one thread per kernel k
    {
        const int k = tid;
        float Sk = 0.f, kmacc[3] = {0.f, 0.f, 0.f};
        for (int c = 0; c < NC; ++c) {
            const float w = kw[k * NC + c];
            Sk += w;
            for (int d = 0; d < 3; ++d) kmacc[d] += w * km[(k * NC + c) * 3 + d];
        }
        const float inv = 1.f / Sk;
        float KMk[3];
        for (int d = 0; d < 3; ++d) KMk[d] = kmacc[d] * inv;

        float kcacc[9];
        for (int t = 0; t < 9; ++t) kcacc[t] = 0.f;
        for (int c = 0; c < NC; ++c) {
            const float* q = &kq[(k * NC + c) * 4];
            const float qn = 1.f / sqrtf(q[0]*q[0] + q[1]*q[1] + q[2]*q[2] + q[3]*q[3]);
            const float w = q[0]*qn, x = q[1]*qn, y = q[2]*qn, z = q[3]*qn;
            float R[3][3];
            R[0][0] = 1.f - 2.f*(y*y + z*z); R[0][1] = 2.f*(x*y - w*z); R[0][2] = 2.f*(x*z + w*y);
            R[1][0] = 2.f*(x*y + w*z); R[1][1] = 1.f - 2.f*(x*x + z*z); R[1][2] = 2.f*(y*z - w*x);
            R[2][0] = 2.f*(x*z - w*y); R[2][1] = 2.f*(y*z + w*x); R[2][2] = 1.f - 2.f*(x*x + y*y);
            const float* s = &ksc[(k * NC + c) * 3];
            float Mx[3][3];
            for (int i = 0; i < 3; ++i)
                for (int j = 0; j < 3; ++j) Mx[i][j] = R[i][j] * s[j];
            float kcov[3][3];
            for (int i = 0; i < 3; ++i)
                for (int j = 0; j < 3; ++j)
                    kcov[i][j] = Mx[i][0]*Mx[j][0] + Mx[i][1]*Mx[j][1] + Mx[i][2]*Mx[j][2];
            float dif[3];
            for (int d = 0; d < 3; ++d) dif[d] = km[(k * NC + c) * 3 + d] - KMk[d];
            const float cwv = kw[k * NC + c];
            for (int i = 0; i < 3; ++i)
                for (int j = 0; j < 3; ++j)
                    kcacc[i * 3 + j] += cwv * (kcov[i][j] + dif[i] * dif[j]);
        }
        S[k] = Sk;
        for (int d = 0; d < 3; ++d) KM[k][d] = KMk[d];
        for (int t = 0; t < 9; ++t) KC[k][t] = kcacc[t] * inv;
    }
    __syncthreads();

    // v0w[j] = sum_k S[k] * W0w[k][j]  (folded rank-1 first layer of w-MLP)
    if (tid < HD) {
        float acc = 0.f;
        for (int k = 0; k < NK; ++k) acc += S[k] * W0w[k * HD + tid];
        ws[tid] = acc;
    }

    // fold the two linear (relu=False) MLPs to (alpha, C[])
    const float* W0s[2] = {W0m, W0c};  const float* b0s[2] = {b0m, b0c};
    const float* W1s[2] = {W1m, W1c};  const float* b1s[2] = {b1m, b1c};
    const float* W2s[2] = {W2m, W2c};  const float* b2s[2] = {b2m, b2c};
    const float* W3s[2] = {W3m, W3c};  const float* b3s[2] = {b3m, b3c};

    for (int which = 0; which < 2; ++which) {
        const float* W0 = W0s[which]; const float* b0 = b0s[which];
        const float* W1 = W1s[which]; const float* b1 = b1s[which];
        const float* W2 = W2s[which]; const float* b2 = b2s[which];
        const float* W3 = W3s[which]; const float* b3 = b3s[which];
        if (tid < HD) {
            float acc = 0.f;
            for (int l = 0; l < HD; ++l) acc += W2[tid * HD + l] * W3[l];
            u2[tid] = acc;
        }
        __syncthreads();
        if (tid < HD) {
            float acc = 0.f;
            for (int l = 0; l < HD; ++l) acc += W1[tid * HD + l] * u2[l];
            u1[tid] = acc;
        }
        __syncthreads();
        {
            float acc = 0.f;
            for (int j = 0; j < HD; ++j) acc += W0[tid * HD + j] * u1[j];
            veff[tid] = acc;
        }
        __syncthreads();
        if (tid == 0) {
            float beff = b3[0];
            for (int j = 0; j < HD; ++j)
                beff += b0[j] * u1[j] + b1[j] * u2[j] + b2[j] * W3[j];
            float alpha = 0.f;
            for (int k = 0; k < NK; ++k) alpha += veff[k];
            const int n     = (which == 0) ? 3 : 9;
            const int base  = (which == 0) ? 32 : 36;
            ws[base] = alpha;
            for (int d = 0; d < n; ++d) {
                float acc = beff;
                for (int k = 0; k < NK; ++k) {
                    // runtime select between the two LDS stat arrays (no
                    // pointer-array of LDS addresses -> no static initializer)
                    const float sv = (which == 0) ? KM[k][d] : KC[k][d];
                    acc += sv * veff[k];
                }
                ws[base + 1 + d] = acc;
            }
        }
        __syncthreads();
    }
}

// ---------------------------------------------------------------------------
// Main kernel: w-MLP via f32 WMMA (16x16x4) + affine m/c + feature copy
// 128 threads = 4 waves; each wave owns a 16-input tile; block covers 64.
// ---------------------------------------------------------------------------
__global__ __launch_bounds__(128) void gm_main(
    const float* __restrict__ weights, const float* __restrict__ means,
    const float* __restrict__ covars,  const float* __restrict__ features,
    const float* __restrict__ b0w,
    const float* __restrict__ W1w, const float* __restrict__ b1w,
    const float* __restrict__ W2w, const float* __restrict__ b2w,
    const float* __restrict__ W3w, const float* __restrict__ b3w,
    const float* __restrict__ ws, float* __restrict__ out)
{
    __shared__ float H[4][16][33];   // per-wave activation tile (16 x 32), padded

    const int tid  = threadIdx.x;
    const int lane = tid & 31;
    const int wave = tid >> 5;
    const int i0   = blockIdx.x * 64;
    const int iBase = i0 + wave * 16;

    // layer 0 (rank-1 folded): H1[m][j] = relu(w[i] * v0[j] + b0[j])
    const float* v0 = ws;
    for (int t = lane; t < 16 * 32; t += 32) {
        const int m = t >> 5, j = t & 31;
        const float h = weights[iBase + m] * v0[j] + b0w[j];
        H[wave][m][j] = h > 0.f ? h : 0.f;
    }
    __syncthreads();

    // layers 1,2: (16x32) @ (32x32) + b, relu — V_WMMA_F32_16X16X4_F32
    const float* Wl[2] = {W1w, W2w};
    const float* bl[2] = {b1w, b2w};
    const int M      = lane & 15;            // A row / B,C column-in-half
    const int kb     = (lane >> 4) << 1;     // K sub-offset: 0 or 2
    const int rowOff = (lane >> 4) << 3;     // C row offset: 0 or 8

    for (int layer = 0; layer < 2; ++layer) {
        // A fragments from LDS (M x K, K split 8 steps of 4)
        float a0[8], a1[8];
        for (int kk = 0; kk < 8; ++kk) {
            a0[kk] = H[wave][M][4 * kk + kb];
            a1[kk] = H[wave][M][4 * kk + kb + 1];
        }
        __syncthreads();

        const float* W = Wl[layer];
        const float* bb = bl[layer];
        v8f acc0, acc1;
        const float bv0 = bb[M], bv1 = bb[16 + M];
        for (int r = 0; r < 8; ++r) { acc0[r] = bv0; acc1[r] = bv1; }

        for (int kk = 0; kk < 8; ++kk) {
            const int r0 = 4 * kk + kb;
            v2f a;  a.x = a0[kk];               a.y = a1[kk];
            v2f f0; f0.x = W[r0 * 32 + M];      f0.y = W[(r0 + 1) * 32 + M];
            v2f f1; f1.x = W[r0 * 32 + 16 + M]; f1.y = W[(r0 + 1) * 32 + 16 + M];
            acc0 = __builtin_amdgcn_wmma_f32_16x16x4_f32(
                false, a, false, f0, (short)0, acc0, false, false);
            acc1 = __builtin_amdgcn_wmma_f32_16x16x4_f32(
                false, a, false, f1, (short)0, acc1, false, false);
        }
        // relu + write back to LDS in plain [m][j] layout
        for (int r = 0; r < 8; ++r) {
            const float x0 = acc0[r], x1 = acc1[r];
            H[wave][rowOff + r][M]      = x0 > 0.f ? x0 : 0.f;
            H[wave][rowOff + r][16 + M] = x1 > 0.f ? x1 : 0.f;
        }
        __syncthreads();
    }

    // final 32 -> 1 (no relu): lanes 0..15, one row each
    if (lane < 16) {
        float acc = b3w[0];
        for (int j = 0; j < 32; ++j) acc += H[wave][lane][j] * W3w[j];
        out[OFF_W + iBase + lane] = acc;
    }

    // affine m_out / c_out + feature passthrough (block-wide, 64 inputs)
    const float alpha_m = ws[32];
    const float alpha_c = ws[36];
    for (int t = tid; t < 64 * 3; t += 128) {
        const int il = t / 3, d = t % 3;
        out[OFF_M + (i0 + il) * 3 + d] =
            alpha_m * means[(i0 + il) * 3 + d] + ws[33 + d];
    }
    for (int t = tid; t < 64 * 9; t += 128) {
        const int il = t / 9, de = t % 9;
        out[OFF_C + (i0 + il) * 9 + de] =
            alpha_c * covars[(i0 + il) * 9 + de] + ws[37 + de];
    }
    const float4* fsrc = (const float4*)(features + (size_t)i0 * NF);
    float4*       fdst = (float4*)(out + OFF_F + (size_t)i0 * NF);
    for (int t = tid; t < 64 * NF / 4; t += 128) fdst[t] = fsrc[t];
}

// ---------------------------------------------------------------------------
// Launch
// ---------------------------------------------------------------------------
extern "C" void kernel_launch(void* const* d_in, const int* in_sizes, int n_in,
                              void* d_out, int out_size, void* d_ws, size_t ws_size,
                              hipStream_t stream) {
    const float* weights = (const float*)d_in[0];
    const float* means   = (const float*)d_in[1];
    const float* covars  = (const float*)d_in[2];
    const float* feats   = (const float*)d_in[3];
    const float* kw      = (const float*)d_in[4];
    const float* km      = (const float*)d_in[5];
    const float* ksc     = (const float*)d_in[6];
    const float* kq      = (const float*)d_in[7];
    // wmlp leaves: 8..15, mmlp: 16..23, cmlp: 24..31
    const float* W0w = (const float*)d_in[8];
    const float* b0w = (const float*)d_in[9];
    const float* W1w = (const float*)d_in[10];
    const float* b1w = (const float*)d_in[11];
    const float* W2w = (const float*)d_in[12];
    const float* b2w = (const float*)d_in[13];
    const float* W3w = (const float*)d_in[14];
    const float* b3w = (const float*)d_in[15];
    const float* W0m = (const float*)d_in[16];
    const float* b0m = (const float*)d_in[17];
    const float* W1m = (const float*)d_in[18];
    const float* b1m = (const float*)d_in[19];
    const float* W2m = (const float*)d_in[20];
    const float* b2m = (const float*)d_in[21];
    const float* W3m = (const float*)d_in[22];
    const float* b3m = (const float*)d_in[23];
    const float* W0c = (const float*)d_in[24];
    const float* b0c = (const float*)d_in[25];
    const float* W1c = (const float*)d_in[26];
    const float* b1c = (const float*)d_in[27];
    const float* W2c = (const float*)d_in[28];
    const float* b2c = (const float*)d_in[29];
    const float* W3c = (const float*)d_in[30];
    const float* b3c = (const float*)d_in[31];

    float* ws  = (float*)d_ws;
    float* out = (float*)d_out;

    gm_precompute<<<1, 64, 0, stream>>>(
        kw, km, ksc, kq, W0w,
        W0m, b0m, W1m, b1m, W2m, b2m, W3m, b3m,
        W0c, b0c, W1c, b1c, W2c, b2c, W3c, b3c, ws);

    gm_main<<<NI / 64, 128, 0, stream>>>(
        weights, means, covars, feats,
        b0w, W1w, b1w, W2w, b2w, W3w, b3w, ws, out);
}